// BitGatConv_48524540510805
// MI455X (gfx1250) — compile-verified
//
#include <hip/hip_runtime.h>
#include <stdint.h>

#define N_NODES 1536
#define IN_CH   256
#define HC      64
#define NEG_SLOPE 0.2f

typedef __attribute__((ext_vector_type(2))) float v2f;
typedef __attribute__((ext_vector_type(8))) float v8f;

// ---------------------------------------------------------------------------
// FP32 WMMA GEMM: C[M,N] = A[M,K] * B[K,N], all row-major, shapes static.
// One wave32 computes one 16x16 tile of C via V_WMMA_F32_16X16X4_F32.
// Grid is sized so every wave maps to a valid tile (EXEC all-ones, no branch).
//
// ISA 7.12.2 layouts (wave32):
//   A (16x4 f32):  lane L -> row M = L&15;  VGPR0/1 hold K = 2*(L>>4) + {0,1}
//   B (4x16 f32):  lane L -> col Nn = L&15; VGPR0/1 hold K = 2*(L>>4) + {0,1}
//   C/D (16x16):   VGPR r, lane L -> D[r + 8*(L>>4)][L&15]
// ---------------------------------------------------------------------------
template <int M, int N, int K>
__global__ __launch_bounds__(128) void wmma_gemm_f32(
    const float* __restrict__ A, const float* __restrict__ B,
    float* __restrict__ C) {
  constexpr int TILES_N = N / 16;
  const int lane = threadIdx.x & 31;
  const int wave = blockIdx.x * (blockDim.x >> 5) + (threadIdx.x >> 5);
  const int tm   = (wave / TILES_N) << 4;   // tile row origin
  const int tn   = (wave % TILES_N) << 4;   // tile col origin
  const int half = lane >> 4;               // 0: K pair {0,1}; 1: {2,3}
  const int l    = lane & 15;               // M index for A, N index for B/C/D

  v8f acc = {};
  const float* arow = A + (size_t)(tm + l) * K;
  #pragma unroll 4
  for (int k = 0; k < K; k += 4) {
    const int ka = k + half * 2;
    v2f a, b;
    a.x = arow[ka];
    a.y = arow[ka + 1];
    b.x = B[(size_t)ka * N + tn + l];
    b.y = B[(size_t)(ka + 1) * N + tn + l];
    // (neg_a, A, neg_b, B, c_mod, C, reuse_a, reuse_b)
    acc = __builtin_amdgcn_wmma_f32_16x16x4_f32(
        false, a, false, b, (short)0, acc, false, false);
  }
  #pragma unroll
  for (int r = 0; r < 8; ++r) {
    C[(size_t)(tm + r + half * 8) * N + tn + l] = acc[r];
  }
}

// ---------------------------------------------------------------------------
// Fused GAT attention: one workgroup (8 waves) per output row i.
//   score[j,c] = leaky_relu(f1[i,c] + f2[j,c]) + bias[i,j]
//   out[i,c]   = nhs[i,c] * (sum_j softmax_j(score))      ( = nhs * S/S )
// Bias row (6 KB) is staged into LDS once via global_load_async_to_lds_b128
// (ASYNCcnt path); both softmax passes then read it as a conflict-free LDS
// broadcast (j is wave-uniform). f2 loads are 128 B/wave coalesced, L2-hot.
// ---------------------------------------------------------------------------
__global__ __launch_bounds__(256) void gat_softmax_fused(
    const float* __restrict__ f1, const float* __restrict__ f2,
    const float* __restrict__ bias, const float* __restrict__ nhs,
    float* __restrict__ out) {
  const int i   = blockIdx.x;
  const int tid = threadIdx.x;
  const int c   = tid & (HC - 1);
  const int jg  = tid >> 6;               // 0..3

  __shared__ float s_bias[N_NODES];       // 6 KB
  __shared__ float s_f1[HC];
  __shared__ float s_red[256];

  const float* __restrict__ brow = bias + (size_t)i * N_NODES;

  // ---- async-stage the bias row into LDS (384 x b128 over 256 lanes) ----
  {
    const uint32_t lds_base = (uint32_t)(uintptr_t)&s_bias[0];
    for (int chunk = tid; chunk < N_NODES / 4; chunk += 256) {
      const unsigned long long ga =
          (unsigned long long)(uintptr_t)(brow + chunk * 4);
      const uint32_t la = lds_base + (uint32_t)chunk * 16u;
      asm volatile("global_load_async_to_lds_b128 %0, %1, off"
                   :: "v"(la), "v"(ga) : "memory");
    }
  }
  if (tid < HC) s_f1[tid] = f1[(size_t)i * HC + tid];
  asm volatile("s_wait_asynccnt 0x0" ::: "memory");
  __syncthreads();

  const float f1c = s_f1[c];

  // ---- pass 1: row max over j ----
  float m = -3.402823466e38f;
  #pragma unroll 4
  for (int j = jg; j < N_NODES; j += 4) {
    float x = f1c + f2[(size_t)j * HC + c];
    x = fmaxf(x, NEG_SLOPE * x);          // leaky_relu, slope in (0,1)
    x += s_bias[j];
    m = fmaxf(m, x);
  }
  s_red[tid] = m;
  __syncthreads();
  const float M = fmaxf(fmaxf(s_red[c], s_red[64 + c]),
                        fmaxf(s_red[128 + c], s_red[192 + c]));
  __syncthreads();

  // ---- pass 2: sum of exp(score - M) ----
  float s = 0.0f;
  #pragma unroll 4
  for (int j = jg; j < N_NODES; j += 4) {
    float x = f1c + f2[(size_t)j * HC + c];
    x = fmaxf(x, NEG_SLOPE * x);
    x += s_bias[j];
    s += __expf(x - M);
  }
  s_red[tid] = s;
  __syncthreads();

  if (tid < HC) {
    const float S = s_red[tid] + s_red[64 + tid] + s_red[128 + tid] +
                    s_red[192 + tid];
    // sum_j coefs = S / S; vals = nhs * that (keeps reference's fp semantics)
    out[(size_t)i * HC + tid] = nhs[(size_t)i * HC + tid] * (S / S);
  }
}

extern "C" void kernel_launch(void* const* d_in, const int* in_sizes, int n_in,
                              void* d_out, int out_size, void* d_ws, size_t ws_size,
                              hipStream_t stream) {
  const float* nodes_ft = (const float*)d_in[0];  // [1536, 256]
  const float* adj_bias = (const float*)d_in[1];  // [1536, 1536]
  const float* weight   = (const float*)d_in[2];  // [256, 64]
  const float* W1       = (const float*)d_in[3];  // [64, 64]
  const float* W2       = (const float*)d_in[4];  // [64, 64]
  float* out = (float*)d_out;                     // [1536, 64]

  float* nhs = (float*)d_ws;                      // [1536, 64]
  float* f1  = nhs + (size_t)N_NODES * HC;        // [1536, 64]
  float* f2  = f1  + (size_t)N_NODES * HC;        // [1536, 64]

  // (1536/16)*(64/16) = 384 waves per GEMM; 4 waves/block -> 96 blocks exact.
  wmma_gemm_f32<N_NODES, HC, IN_CH><<<96, 128, 0, stream>>>(nodes_ft, weight, nhs);
  wmma_gemm_f32<N_NODES, HC, HC><<<96, 128, 0, stream>>>(nhs, W1, f1);
  wmma_gemm_f32<N_NODES, HC, HC><<<96, 128, 0, stream>>>(nhs, W2, f2);

  gat_softmax_fused<<<N_NODES, 256, 0, stream>>>(f1, f2, adj_bias, nhs, out);
}